// RCOptimModel_69733089017916
// MI455X (gfx1250) — compile-verified
//
#include <hip/hip_runtime.h>
#include <math.h>

// ---------------- problem constants ----------------
#define NST 256        // states
#define NROOMS 254
#define MCOLS 255      // inputs (1 Tout + 254 rooms)
#define TSTEPS 100000
#define K2 125         // chunk length (TSTEPS = 800 * 125)
#define NCH 800
#define GCH 16         // chunks batched per interior workgroup
#define NBLK (NCH / GCH)   // 50 interior blocks
#define DTF 30.0f
#define OMEGA 7.2722052166430399e-05f   // 2*pi / 86400

#define LDS_PHIT (257*256)   // padded transposed 256x256 matrix (scan/buildU/buildH)
#define PHL 260              // row stride (floats) of row-major padded Phi for WMMA A-frags
#define XL 18                // row stride (floats) of X block in LDS (k-major, 16 cols)

typedef __attribute__((ext_vector_type(2))) float v2f;
typedef __attribute__((ext_vector_type(8))) float v8f;

// ---------------- elementwise helpers ----------------
__global__ void k_scale(const float* __restrict__ A, float* __restrict__ O, float s, int n) {
  int i = blockIdx.x * blockDim.x + threadIdx.x;
  if (i < n) O[i] = s * A[i];
}

// Phi = I + M + M^2/2 + M^3/6 + M^4/24
__global__ void k_phi(const float* __restrict__ M1, const float* __restrict__ M2,
                      const float* __restrict__ M3, const float* __restrict__ M4,
                      float* __restrict__ Phi) {
  int i = blockIdx.x * blockDim.x + threadIdx.x;
  if (i >= NST * NST) return;
  int r = i >> 8, c = i & 255;
  Phi[i] = (r == c ? 1.0f : 0.0f) + M1[i] + 0.5f * M2[i]
         + (1.0f / 6.0f) * M3[i] + (1.0f / 24.0f) * M4[i];
}

// ---------------- 256x256x256 f32 GEMM via V_WMMA_F32_16X16X4_F32 ----------------
// One wave32 per 16x16 output tile; K-loop of 64 wmma ops (K=4 each).
// A frag (16x4): lane l -> M=l%16, K=kb*4+2*(l/16)+{0,1}
// B frag (4x16): lane l -> N=l%16, K=kb*4+2*(l/16)+{0,1}
// C/D (16x16):   vgpr v, lane l -> M=v+8*(l/16), N=l%16
__global__ void k_gemm256(const float* __restrict__ A, const float* __restrict__ B,
                          float* __restrict__ C) {
  const int lane = threadIdx.x;              // 0..31 (wave32)
  const int ti = blockIdx.x >> 4;
  const int tj = blockIdx.x & 15;
  const int mrow = lane & 15;
  const int half = lane >> 4;
  const int arow = (ti * 16 + mrow) * NST;
  const int bcol = tj * 16 + mrow;
  v8f c = {};
#pragma unroll 4
  for (int kb = 0; kb < 64; ++kb) {
    const int k0 = kb * 4 + 2 * half;
    v2f a, b;
    a.x = A[arow + k0];
    a.y = A[arow + k0 + 1];
    b.x = B[k0 * NST + bcol];
    b.y = B[(k0 + 1) * NST + bcol];
    c = __builtin_amdgcn_wmma_f32_16x16x4_f32(false, a, false, b, (short)0, c,
                                              false, false);
  }
#pragma unroll
  for (int v = 0; v < 8; ++v)
    C[(ti * 16 + v + 8 * half) * NST + tj * 16 + mrow] = c[v];
}

// ---------------- scalar streams s_n = [TV[n], TV[n+1], 1, sin(w t), cos(w t)] ----------------
__global__ void k_streams(const float* __restrict__ TV, float* __restrict__ S) {
  int n = blockIdx.x * blockDim.x + threadIdx.x;
  if (n >= TSTEPS) return;
  float t = (float)n * DTF;
  float ang = OMEGA * t;
  S[0 * TSTEPS + n] = TV[n];
  S[1 * TSTEPS + n] = TV[n + 1];
  S[2 * TSTEPS + n] = 1.0f;
  S[3 * TSTEPS + n] = sinf(ang);
  S[4 * TSTEPS + n] = cosf(ang);
}

// ---------------- build U (256x5) ----------------
// D_i = (dt/8) * poly_i(M) * B, poly1=I+M+M^2/3+M^3/3, poly2=3I+2M+M^2,
// poly3=3I+M, poly4=I. Rank-5 forcing basis (see derivation in comments above).
__global__ void k_buildU(const float* __restrict__ Mg, const float* __restrict__ Bg,
                         const float* __restrict__ heating, float* __restrict__ U) {
  extern __shared__ float lds[];
  float* mt  = lds;                   // 257*256 transposed-padded M
  float* qv  = mt + LDS_PHIT;         // 254
  float* phs = qv + NROOMS;           // 254
  float* wc  = phs + NROOMS;          // 4*254
  float* ws  = wc + 4 * NROOMS;       // 4*254
  float* B0L = ws + 4 * NROOMS;       // 256
  float* BQL = B0L + NST;             // 256
  float* BC  = BQL + NST;             // 4*256
  float* BS  = BC + 4 * NST;          // 4*256
  float* V1  = BS + 4 * NST;          // 256
  float* V2  = V1 + NST;              // 256
  float* V3  = V2 + NST;              // 256
  const int tid = threadIdx.x;

  for (int c = 0; c < NST; ++c) mt[tid * 257 + c] = Mg[c * NST + tid];
  if (tid < NROOMS) {
    float h0 = heating[tid * 3 + 0];
    float h1 = heating[tid * 3 + 1];
    float h2 = heating[tid * 3 + 2];
    float qavg = 500.0f / (1.0f + expf(-h0));
    float amp = 1.0f / (1.0f + expf(-h2));
    qv[tid] = qavg * amp;
    phs[tid] = h1;
  }
  __syncthreads();
  for (int idx = tid; idx < 4 * NROOMS; idx += NST) {
    int i = idx / NROOMS, j = idx - i * NROOMS;   // i=0..3 -> delta = 10*i seconds
    float ang = OMEGA * (10.0f * (float)i) + phs[j];
    wc[idx] = qv[j] * cosf(ang);
    ws[idx] = qv[j] * sinf(ang);
  }
  __syncthreads();
  {
    float b0 = Bg[tid * MCOLS + 0];
    float aq = 0.f;
    float ac0 = 0.f, ac1 = 0.f, ac2 = 0.f, ac3 = 0.f;
    float as0 = 0.f, as1 = 0.f, as2 = 0.f, as3 = 0.f;
    for (int j = 0; j < NROOMS; ++j) {
      float bj = Bg[tid * MCOLS + 1 + j];
      aq  += bj * qv[j];
      ac0 += bj * wc[0 * NROOMS + j];
      ac1 += bj * wc[1 * NROOMS + j];
      ac2 += bj * wc[2 * NROOMS + j];
      ac3 += bj * wc[3 * NROOMS + j];
      as0 += bj * ws[0 * NROOMS + j];
      as1 += bj * ws[1 * NROOMS + j];
      as2 += bj * ws[2 * NROOMS + j];
      as3 += bj * ws[3 * NROOMS + j];
    }
    B0L[tid] = b0; BQL[tid] = aq;
    BC[0 * NST + tid] = ac0; BC[1 * NST + tid] = ac1;
    BC[2 * NST + tid] = ac2; BC[3 * NST + tid] = ac3;
    BS[0 * NST + tid] = as0; BS[1 * NST + tid] = as1;
    BS[2 * NST + tid] = as2; BS[3 * NST + tid] = as3;
  }

#define MV(dst, src)                                            \
  {                                                             \
    __syncthreads();                                            \
    float a_ = 0.f;                                             \
    for (int k_ = 0; k_ < NST; ++k_)                            \
      a_ += mt[k_ * 257 + tid] * (src)[k_];                     \
    __syncthreads();                                            \
    (dst)[tid] = a_;                                            \
  }

  const float c8 = DTF / 8.0f, c16 = DTF / 16.0f;
  float p_r, q_r, cst_r, vc_r, vs_r;

  MV(V1, B0L); MV(V2, V1); MV(V3, V2);
  p_r = c8 * (4.0f * B0L[tid] + (8.0f / 3.0f) * V1[tid] + V2[tid] + (1.0f / 3.0f) * V3[tid]);
  q_r = c8 * (4.0f * B0L[tid] + (4.0f / 3.0f) * V1[tid] + (1.0f / 3.0f) * V2[tid]);

  MV(V1, BQL); MV(V2, V1); MV(V3, V2);
  cst_r = c16 * (8.0f * BQL[tid] + 4.0f * V1[tid] + (4.0f / 3.0f) * V2[tid] + (1.0f / 3.0f) * V3[tid]);

  MV(V1, BC + 0 * NST); MV(V2, V1); MV(V3, V2);
  vc_r = c16 * (BC[0 * NST + tid] + V1[tid] + (1.0f / 3.0f) * V2[tid] + (1.0f / 3.0f) * V3[tid]);
  MV(V1, BC + 1 * NST); MV(V2, V1);
  vc_r += c16 * (3.0f * BC[1 * NST + tid] + 2.0f * V1[tid] + V2[tid]);
  MV(V1, BC + 2 * NST);
  vc_r += c16 * (3.0f * BC[2 * NST + tid] + V1[tid]);
  vc_r += c16 * BC[3 * NST + tid];

  MV(V1, BS + 0 * NST); MV(V2, V1); MV(V3, V2);
  vs_r = c16 * (BS[0 * NST + tid] + V1[tid] + (1.0f / 3.0f) * V2[tid] + (1.0f / 3.0f) * V3[tid]);
  MV(V1, BS + 1 * NST); MV(V2, V1);
  vs_r += c16 * (3.0f * BS[1 * NST + tid] + 2.0f * V1[tid] + V2[tid]);
  MV(V1, BS + 2 * NST);
  vs_r += c16 * (3.0f * BS[2 * NST + tid] + V1[tid]);
  vs_r += c16 * BS[3 * NST + tid];
#undef MV

  U[tid * 5 + 0] = p_r;
  U[tid * 5 + 1] = q_r;
  U[tid * 5 + 2] = cst_r;
  U[tid * 5 + 3] = vc_r;
  U[tid * 5 + 4] = vs_r;
}

// ---------------- H_l = Phi^l U, l = 0..K2-1, layout Hcat[(l*5+c)*256 + r] ----------------
__global__ void k_buildH(const float* __restrict__ Phi, const float* __restrict__ U,
                         float* __restrict__ Hcat) {
  extern __shared__ float lds[];
  float* pt = lds;                  // 257*256
  float* hcur = pt + LDS_PHIT;      // 1280
  const int tid = threadIdx.x;
  for (int c = 0; c < NST; ++c) pt[tid * 257 + c] = Phi[c * NST + tid];
  for (int c = 0; c < 5; ++c) {
    float v = U[tid * 5 + c];
    hcur[tid * 5 + c] = v;
    Hcat[(0 * 5 + c) * NST + tid] = v;
  }
  __syncthreads();
  for (int l = 1; l < K2; ++l) {
    float a0 = 0.f, a1 = 0.f, a2 = 0.f, a3 = 0.f, a4 = 0.f;
    for (int k = 0; k < NST; ++k) {
      float pv = pt[k * 257 + tid];
      a0 += pv * hcur[k * 5 + 0];
      a1 += pv * hcur[k * 5 + 1];
      a2 += pv * hcur[k * 5 + 2];
      a3 += pv * hcur[k * 5 + 3];
      a4 += pv * hcur[k * 5 + 4];
    }
    __syncthreads();
    hcur[tid * 5 + 0] = a0; hcur[tid * 5 + 1] = a1; hcur[tid * 5 + 2] = a2;
    hcur[tid * 5 + 3] = a3; hcur[tid * 5 + 4] = a4;
    Hcat[(l * 5 + 0) * NST + tid] = a0;
    Hcat[(l * 5 + 1) * NST + tid] = a1;
    Hcat[(l * 5 + 2) * NST + tid] = a2;
    Hcat[(l * 5 + 3) * NST + tid] = a3;
    Hcat[(l * 5 + 4) * NST + tid] = a4;
    __syncthreads();
  }
}

// ---------------- chunk carries r2_m = sum_{j<K2} Phi^(K2-1-j) U s_{m*K2+j} ----------------
__global__ void k_r2(const float* __restrict__ Hcat, const float* __restrict__ S,
                     float* __restrict__ R2) {
  __shared__ float sl[K2 * 5];
  const int m = blockIdx.x, tid = threadIdx.x;
  for (int idx = tid; idx < K2 * 5; idx += NST) {
    int l = idx / 5, c = idx - 5 * l;
    int n = m * K2 + (K2 - 1 - l);
    sl[idx] = S[c * TSTEPS + n];
  }
  __syncthreads();
  float acc = 0.f;
  for (int idx = 0; idx < K2 * 5; ++idx) acc += Hcat[idx * NST + tid] * sl[idx];
  R2[m * NST + tid] = acc;
}

// ---------------- serial boundary scan: z_{m+1} = Phi^K2 z_m + r2_m ----------------
__global__ void k_scan(const float* __restrict__ PhiK, const float* __restrict__ R2,
                       const float* __restrict__ iv, float* __restrict__ Z,
                       float* __restrict__ out) {
  extern __shared__ float lds[];
  float* pt = lds;
  float* x = pt + LDS_PHIT;
  const int tid = threadIdx.x;
  for (int c = 0; c < NST; ++c) pt[tid * 257 + c] = PhiK[c * NST + tid];
  float xv = iv[tid];
  x[tid] = xv;
  Z[tid] = xv;
  out[tid] = xv;                         // out[0] = x_0 = iv
  __syncthreads();
  for (int m = 0; m < NCH; ++m) {
    float acc = R2[m * NST + tid];
#pragma unroll 8
    for (int k = 0; k < NST; ++k) acc += pt[k * 257 + tid] * x[k];
    __syncthreads();
    x[tid] = acc;
    Z[(size_t)(m + 1) * NST + tid] = acc;
    out[(size_t)(m + 1) * K2 * NST + tid] = acc;   // boundary states
    __syncthreads();
  }
}

// ---------------- interior reconstruction on the matrix pipe ----------------
// Each block replays GCH=16 chunks simultaneously: X(256x16) <- Phi(256x256) @ X + U s_n,
// where column j of X is chunk m0+j at local step d. The rank-5 forcing U*s is folded
// into the WMMA accumulator init (C operand). Phi is LDS-resident, row-major padded to
// PHL floats (conflict-free b64 A-frag reads). X is LDS-resident, k-major with XL pad
// (conflict-free B-frag reads and D write-back). 8 waves x 2 row-tiles = 256 rows.
__global__ void k_interior(const float* __restrict__ Phi, const float* __restrict__ U,
                           const float* __restrict__ S, const float* __restrict__ Z,
                           float* __restrict__ out) {
  extern __shared__ float lds[];
  float* phl = lds;                 // PHL*256 row-major padded Phi
  float* Xl  = phl + PHL * NST;     // 256*XL state block
  const int tid = threadIdx.x;
  const int lane = tid & 31;
  const int wave = tid >> 5;        // 0..7
  const int col  = lane & 15;       // N index / chunk column
  const int half = lane >> 4;       // K/M split
  const int ti0 = 2 * wave, ti1 = 2 * wave + 1;
  const int m0 = blockIdx.x * GCH;

  // fill padded Phi (coalesced global reads)
  for (int idx = tid; idx < NST * NST; idx += NST)
    phl[(idx >> 8) * PHL + (idx & 255)] = Phi[idx];
  // init X columns from boundary states
  for (int c = 0; c < GCH; ++c)
    Xl[tid * XL + c] = Z[(size_t)(m0 + c) * NST + tid];

  // preload per-lane forcing rows: rows covered by this lane's D fragments
  float ua[5][8], ub[5][8];
#pragma unroll
  for (int v = 0; v < 8; ++v) {
    const int r0 = (ti0 * 16 + 8 * half + v) * 5;
    const int r1 = (ti1 * 16 + 8 * half + v) * 5;
#pragma unroll
    for (int c = 0; c < 5; ++c) { ua[c][v] = U[r0 + c]; ub[c][v] = U[r1 + c]; }
  }
  __syncthreads();

  for (int d = 0; d < K2 - 1; ++d) {          // d = K2-1 state is written by the scan
    const int ncol = (m0 + col) * K2 + d;     // this column's global step index
    const float s0 = S[0 * TSTEPS + ncol];
    const float s1 = S[1 * TSTEPS + ncol];
    const float s3 = S[3 * TSTEPS + ncol];
    const float s4 = S[4 * TSTEPS + ncol];

    v8f acc0, acc1;                            // C = U s_n (rank-5 forcing)
#pragma unroll
    for (int v = 0; v < 8; ++v) {
      acc0[v] = ua[0][v] * s0 + ua[1][v] * s1 + ua[2][v] + ua[3][v] * s3 + ua[4][v] * s4;
      acc1[v] = ub[0][v] * s0 + ub[1][v] * s1 + ub[2][v] + ub[3][v] * s3 + ub[4][v] * s4;
    }
#pragma unroll 4
    for (int kb = 0; kb < 64; ++kb) {
      const int kbase = kb * 4 + 2 * half;
      v2f b, a0, a1;
      b.x = Xl[kbase * XL + col];
      b.y = Xl[(kbase + 1) * XL + col];
      a0.x = phl[(ti0 * 16 + col) * PHL + kbase];
      a0.y = phl[(ti0 * 16 + col) * PHL + kbase + 1];
      a1.x = phl[(ti1 * 16 + col) * PHL + kbase];
      a1.y = phl[(ti1 * 16 + col) * PHL + kbase + 1];
      acc0 = __builtin_amdgcn_wmma_f32_16x16x4_f32(false, a0, false, b, (short)0,
                                                   acc0, false, false);
      acc1 = __builtin_amdgcn_wmma_f32_16x16x4_f32(false, a1, false, b, (short)0,
                                                   acc1, false, false);
    }
    __syncthreads();                           // all B reads done before overwrite
#pragma unroll
    for (int v = 0; v < 8; ++v) {
      Xl[(ti0 * 16 + v + 8 * half) * XL + col] = acc0[v];
      Xl[(ti1 * 16 + v + 8 * half) * XL + col] = acc1[v];
    }
    __syncthreads();
    // coalesced column stores: out[n+1][:] for each chunk column
#pragma unroll
    for (int c = 0; c < GCH; ++c)
      out[(size_t)((m0 + c) * K2 + d + 1) * NST + tid] = Xl[tid * XL + c];
  }
}

// ---------------- host launcher ----------------
extern "C" void kernel_launch(void* const* d_in, const int* in_sizes, int n_in,
                              void* d_out, int out_size, void* d_ws, size_t ws_size,
                              hipStream_t stream) {
  const float* A  = (const float*)d_in[0];
  const float* B  = (const float*)d_in[1];
  const float* heating = (const float*)d_in[2];
  const float* TV = (const float*)d_in[4];   // Tout_values (T+1)
  const float* iv = (const float*)d_in[5];
  float* out = (float*)d_out;
  float* ws = (float*)d_ws;

  const int NN = NST * NST;
  float* M1   = ws;            // dt*A
  float* M2   = M1 + NN;
  float* M3   = M2 + NN;
  float* M4   = M3 + NN;
  float* Phi  = M4 + NN;
  float* T1   = Phi + NN;
  float* T2   = T1 + NN;
  float* Q1   = T2 + NN;
  float* Q2   = Q1 + NN;
  float* PhiK = Q2 + NN;
  float* U    = PhiK + NN;               // 256*5
  float* S    = U + NST * 5;             // 5*TSTEPS
  float* Hcat = S + 5 * TSTEPS;          // K2*5*256
  float* R2   = Hcat + K2 * 5 * NST;     // NCH*256
  float* Z    = R2 + NCH * NST;          // (NCH+1)*256

  k_scale<<<(NN + 255) / 256, 256, 0, stream>>>(A, M1, DTF, NN);
  k_gemm256<<<256, 32, 0, stream>>>(M1, M1, M2);
  k_gemm256<<<256, 32, 0, stream>>>(M2, M1, M3);
  k_gemm256<<<256, 32, 0, stream>>>(M2, M2, M4);
  k_phi<<<(NN + 255) / 256, 256, 0, stream>>>(M1, M2, M3, M4, Phi);

  // PhiK = Phi^125 = ((Phi^5)^5)^5; pow5: in^2 -> in^4 -> in^5
  auto pow5 = [&](const float* in, float* o) {
    k_gemm256<<<256, 32, 0, stream>>>(in, in, T1);
    k_gemm256<<<256, 32, 0, stream>>>(T1, T1, T2);
    k_gemm256<<<256, 32, 0, stream>>>(T2, in, o);
  };
  pow5(Phi, Q1);
  pow5(Q1, Q2);
  pow5(Q2, PhiK);

  k_streams<<<(TSTEPS + 255) / 256, 256, 0, stream>>>(TV, S);

  size_t ldsU = (size_t)(LDS_PHIT + 2 * NROOMS + 8 * NROOMS + 10 * NST + 3 * NST) * sizeof(float);
  k_buildU<<<1, 256, ldsU, stream>>>(M1, B, heating, U);

  size_t ldsH = (size_t)(LDS_PHIT + 5 * NST) * sizeof(float);
  k_buildH<<<1, 256, ldsH, stream>>>(Phi, U, Hcat);

  k_r2<<<NCH, 256, 0, stream>>>(Hcat, S, R2);

  size_t ldsS = (size_t)(LDS_PHIT + NST) * sizeof(float);
  k_scan<<<1, 256, ldsS, stream>>>(PhiK, R2, iv, Z, out);

  size_t ldsI = (size_t)(PHL * NST + NST * XL) * sizeof(float);
  k_interior<<<NBLK, 256, ldsI, stream>>>(Phi, U, S, Z, out);
}